// diff_rc2lpc_52664888984119
// MI455X (gfx1250) — compile-verified
//
#include <hip/hip_runtime.h>

// rc2lpc: reflection coefficients -> LPC coefficients.
// x: (B=128, T=16000, N=16) fp32. One thread per (b,t) position.
// Memory-bound streaming kernel: 262 MB total traffic, ~11.3 us floor at 23.3 TB/s.
// Non-temporal (TH=NT) 128-bit loads/stores: every byte is touched exactly once,
// so we avoid polluting the 192 MB L2 and let stores rinse straight to HBM.

#ifndef LPC_N
#define LPC_N 16
#endif

// clang ext_vector type: required by __builtin_nontemporal_load/store
typedef float v4f __attribute__((ext_vector_type(4)));

__global__ __launch_bounds__(256) void rc2lpc_kernel(const float* __restrict__ x,
                                                     float* __restrict__ out,
                                                     int n_problems) {
    int idx = blockIdx.x * blockDim.x + threadIdx.x;
    if (idx >= n_problems) return;

    size_t base = (size_t)idx * LPC_N;

    // ---- load 16 reflection coefficients as 4x 128-bit, non-temporal ----
    const v4f* __restrict__ xp = (const v4f*)(x + base);
    v4f v0 = __builtin_nontemporal_load(xp + 0);
    v4f v1 = __builtin_nontemporal_load(xp + 1);
    v4f v2 = __builtin_nontemporal_load(xp + 2);
    v4f v3 = __builtin_nontemporal_load(xp + 3);

    float r[LPC_N];
    r[0]  = v0.x; r[1]  = v0.y; r[2]  = v0.z; r[3]  = v0.w;
    r[4]  = v1.x; r[5]  = v1.y; r[6]  = v1.z; r[7]  = v1.w;
    r[8]  = v2.x; r[9]  = v2.y; r[10] = v2.z; r[11] = v2.w;
    r[12] = v3.x; r[13] = v3.y; r[14] = v3.z; r[15] = v3.w;

    // ---- Levinson-style recursion, fully unrolled in registers ----
    // step i: new[j] = lpc[j] + k * lpc[i-1-j]  (j < i, old values), new[i] = k
    float lpc[LPC_N];
    lpc[0] = r[0];
#pragma unroll
    for (int i = 1; i < LPC_N; ++i) {
        float k = r[i];
        // pairwise so both reads see pre-update values
#pragma unroll
        for (int j = 0; j < i / 2; ++j) {
            float a = lpc[j];
            float b = lpc[i - 1 - j];
            lpc[j]         = a + k * b;
            lpc[i - 1 - j] = b + k * a;
        }
        if (i & 1) {
            int m = (i - 1) / 2;           // middle element: j == i-1-j
            float a = lpc[m];
            lpc[m] = a + k * a;
        }
        lpc[i] = k;
    }

    // ---- store 16 LPC coefficients as 4x 128-bit, non-temporal ----
    v4f o0 = { lpc[0],  lpc[1],  lpc[2],  lpc[3]  };
    v4f o1 = { lpc[4],  lpc[5],  lpc[6],  lpc[7]  };
    v4f o2 = { lpc[8],  lpc[9],  lpc[10], lpc[11] };
    v4f o3 = { lpc[12], lpc[13], lpc[14], lpc[15] };

    v4f* __restrict__ op = (v4f*)(out + base);
    __builtin_nontemporal_store(o0, op + 0);
    __builtin_nontemporal_store(o1, op + 1);
    __builtin_nontemporal_store(o2, op + 2);
    __builtin_nontemporal_store(o3, op + 3);
}

extern "C" void kernel_launch(void* const* d_in, const int* in_sizes, int n_in,
                              void* d_out, int out_size, void* d_ws, size_t ws_size,
                              hipStream_t stream) {
    const float* x = (const float*)d_in[0];
    // d_in[1] is lpcoeffs_N (==16); it determines output shape so it is baked in as LPC_N.
    float* out = (float*)d_out;

    int n_problems = in_sizes[0] / LPC_N;   // 128*16000 = 2,048,000
    const int block = 256;                   // 8 waves (wave32) per block
    int grid = (n_problems + block - 1) / block;

    rc2lpc_kernel<<<grid, block, 0, stream>>>(x, out, n_problems);
}